// TiSASRec_20959440404899
// MI455X (gfx1250) — compile-verified
//
#include <hip/hip_runtime.h>
#include <hip/hip_bf16.h>

// TiSASRec attention, MI455X (gfx1250). B=8 L=256 H=128 NH=4 D=32.
// Roofline: dominated by one-shot streaming of time_matrix_K/V (537 MB fp32).
// WMMA f32 16x16x4 used for projections and Q*Keff^T; A*Veff fused into the
// mandatory tV stream.

#define B_  8
#define L_  256
#define H_  128
#define NH_ 4
#define D_  32

typedef __attribute__((ext_vector_type(2))) float v2f;
typedef __attribute__((ext_vector_type(8))) float v8f;

__device__ __forceinline__ v8f wmma_f32_4(v2f a, v2f b, v8f c) {
  // (neg_a, A, neg_b, B, c_mod, C, reuse_a, reuse_b)
  return __builtin_amdgcn_wmma_f32_16x16x4_f32(false, a, false, b, (short)0, c,
                                               false, false);
}

// ---------------------------------------------------------------------------
// Projection: out = X @ W^T + bias (+ abs_pos fold for K and V).
// One wave per 16x16 output tile; K-loop of 32 WMMA f32 16x16x4 steps.
// Tile ids: [0,1024)=Q, [1024,2048)=Keff, [2048,3072)=Veff.
// ---------------------------------------------------------------------------
__global__ void __launch_bounds__(128)
proj_kernel(const float* __restrict__ queries, const float* __restrict__ keys,
            const float* __restrict__ Wq, const float* __restrict__ bq,
            const float* __restrict__ Wk, const float* __restrict__ bk,
            const float* __restrict__ Wv, const float* __restrict__ bv,
            const float* __restrict__ pK, const float* __restrict__ pV,
            float* __restrict__ Qb, float* __restrict__ Kb,
            float* __restrict__ Vb) {
  const int g    = blockIdx.x * 4 + (threadIdx.x >> 5);
  const int lane = threadIdx.x & 31;
  const int which = g >> 10;           // 0=Q 1=K 2=V (uniform per wave)
  const int rem   = g & 1023;
  const int r0    = (rem >> 3) * 16;   // row tile in [0, B*L)
  const int c0    = (rem & 7) * 16;    // col tile in [0, H)

  const float* X    = (which == 0) ? queries : keys;
  const float* W    = (which == 0) ? Wq : (which == 1) ? Wk : Wv;
  const float* bias = (which == 0) ? bq : (which == 1) ? bk : bv;
  const float* pos  = (which == 0) ? (const float*)nullptr
                                   : (which == 1) ? pK : pV;
  float* out = (which == 0) ? Qb : (which == 1) ? Kb : Vb;

  const int m  = lane & 15;   // A row / B col index
  const int hi = lane >> 4;   // lane-half select

  const float* xrow = X + (size_t)(r0 + m) * H_;  // A: X rows
  const float* wrow = W + (size_t)(c0 + m) * H_;  // B: W^T cols == W rows

  v8f c = {0.f, 0.f, 0.f, 0.f, 0.f, 0.f, 0.f, 0.f};
  for (int k0 = 0; k0 < H_; k0 += 4) {
    v2f a, b;
    a.x = xrow[k0 + 2 * hi];
    a.y = xrow[k0 + 2 * hi + 1];
    b.x = wrow[k0 + 2 * hi];
    b.y = wrow[k0 + 2 * hi + 1];
    c = wmma_f32_4(a, b, c);
  }
#pragma unroll
  for (int j = 0; j < 8; ++j) {
    const int row = r0 + hi * 8 + j;  // C[8*hi+j][n]
    const int col = c0 + m;
    float v = c[j] + bias[col];
    if (pos) v += pos[(size_t)row * H_ + col];
    out[(size_t)row * H_ + col] = v;
  }
}

// ---------------------------------------------------------------------------
// Fused attention. One 128-thread block per (b, h, 16-row query tile).
//   S = Q*Keff^T (WMMA) + tK-dot bias; mask; softmax; out = A*(Veff + tV).
// ---------------------------------------------------------------------------
__global__ void __launch_bounds__(128)
attn_kernel(const float* __restrict__ Qb, const float* __restrict__ Kb,
            const float* __restrict__ Vb, const float* __restrict__ tK,
            const float* __restrict__ tV,
            const unsigned char* __restrict__ tmask,
            float* __restrict__ out) {
  __shared__ float Qs[16][36];       // padded stride (36*4 bytes, 16B aligned)
  __shared__ float Ks[256][36];
  __shared__ float Ssc[16][257];     // scores -> exp weights

  const int t  = threadIdx.x;
  const int bt = blockIdx.x;
  const int lt = bt & 15;            // query row-tile
  const int h  = (bt >> 4) & 3;      // head
  const int b  = bt >> 6;            // batch
  const int l0 = lt * 16;
  const int hc = h * D_;

  // ---- stage Q tile (16x32) ----
  {
    const int row = t >> 3, c4 = (t & 7) * 4;
    float4 q = *(const float4*)(Qb + (size_t)(b * L_ + l0 + row) * H_ + hc + c4);
    *(float4*)&Qs[row][c4] = q;
  }
  // ---- stage Keff head slice (256x32) ----
  for (int i = 0; i < 16; ++i) {
    const int idx = t + i * 128;
    const int row = idx >> 3, c4 = (idx & 7) * 4;
    float4 k = *(const float4*)(Kb + (size_t)(b * L_ + row) * H_ + hc + c4);
    *(float4*)&Ks[row][c4] = k;
  }
  __syncthreads();

  // ---- S = Q * Keff^T via WMMA f32 16x16x4 (causal tiles only) ----
  {
    const int lane = t & 31;
    const int w    = t >> 5;
    const int mi   = lane & 15;
    const int hi   = lane >> 4;
    for (int mt = w; mt <= lt; mt += 4) {
      v8f c = {0.f, 0.f, 0.f, 0.f, 0.f, 0.f, 0.f, 0.f};
#pragma unroll
      for (int k0 = 0; k0 < D_; k0 += 4) {
        v2f a, bb;
        a.x  = Qs[mi][k0 + 2 * hi];
        a.y  = Qs[mi][k0 + 2 * hi + 1];
        bb.x = Ks[mt * 16 + mi][k0 + 2 * hi];   // B[kk][n] = Keff[m0+n][kk]
        bb.y = Ks[mt * 16 + mi][k0 + 2 * hi + 1];
        c = wmma_f32_4(a, bb, c);
      }
#pragma unroll
      for (int j = 0; j < 8; ++j)
        Ssc[hi * 8 + j][mt * 16 + mi] = c[j];
    }
  }
  __syncthreads();

  // ---- time-interval bias (tK stream), mask, scale ----
  const float scale = 0.17677669529663687f;   // 1/sqrt(32)
  const float NEGV  = -4294967295.0f;         // -(2^32)+1
  for (int i = 0; i < 32; ++i) {
    const int idx  = t + i * 128;
    const int l    = idx >> 8;
    const int mcol = idx & 255;
    const int gl   = l0 + l;
    const bool tm  = tmask[b * L_ + gl] != 0;
    float val;
    if (tm || mcol > gl) {
      val = NEGV;
    } else {
      const float* tk =
          tK + (((size_t)b * L_ + gl) * L_ + mcol) * H_ + hc;  // 128B line
      float dot = 0.f;
#pragma unroll
      for (int d4 = 0; d4 < D_; d4 += 4) {
        float4 tkv = *(const float4*)(tk + d4);
        float4 qv  = *(const float4*)&Qs[l][d4];
        dot += tkv.x * qv.x + tkv.y * qv.y + tkv.z * qv.z + tkv.w * qv.w;
      }
      val = (Ssc[l][mcol] + dot) * scale;
    }
    Ssc[l][mcol] = val;
  }
  __syncthreads();

  // ---- row softmax: 8 lanes per row, shfl_xor reductions (wave32) ----
  const int row = t >> 3, sub = t & 7;
  const int gl  = l0 + row;
  float mx = -3.4e38f;
  for (int j = 0; j < 32; ++j) mx = fmaxf(mx, Ssc[row][sub * 32 + j]);
  mx = fmaxf(mx, __shfl_xor(mx, 1));
  mx = fmaxf(mx, __shfl_xor(mx, 2));
  mx = fmaxf(mx, __shfl_xor(mx, 4));
  float sm = 0.f;
  for (int j = 0; j < 32; ++j) {
    float e = __expf(Ssc[row][sub * 32 + j] - mx);
    Ssc[row][sub * 32 + j] = e;
    sm += e;
  }
  sm += __shfl_xor(sm, 1);
  sm += __shfl_xor(sm, 2);
  sm += __shfl_xor(sm, 4);
  const float inv = 1.0f / sm;
  __syncthreads();

  // ---- out = inv * sum_m A[l,m] * (Veff[m] + tV[l,m])  (tV stream) ----
  // Fully time-masked rows are uniform over ALL 256 keys (matches reference
  // softmax of an all-NEG row); causal rows stop at m = gl.
  const bool tm    = tmask[b * L_ + gl] != 0;
  const int  limit = tm ? L_ : (gl + 1);
  const int  d0    = sub * 4;
  const float* tv  = tV + ((size_t)b * L_ + gl) * L_ * H_ + hc + d0;
  const float* vb  = Vb + (size_t)b * L_ * H_ + hc + d0;
  float ax = 0.f, ay = 0.f, az = 0.f, aw = 0.f;
  for (int mcol = 0; mcol < limit; ++mcol) {
    __builtin_prefetch((const void*)(tv + (size_t)(mcol + 20) * H_), 0, 1);
    const float a = Ssc[row][mcol];
    float4 w1 = *(const float4*)(tv + (size_t)mcol * H_);
    float4 w2 = *(const float4*)(vb + (size_t)mcol * H_);
    ax = fmaf(a, w1.x + w2.x, ax);
    ay = fmaf(a, w1.y + w2.y, ay);
    az = fmaf(a, w1.z + w2.z, az);
    aw = fmaf(a, w1.w + w2.w, aw);
  }
  float4 r;
  r.x = ax * inv; r.y = ay * inv; r.z = az * inv; r.w = aw * inv;
  *(float4*)(out + (size_t)(b * L_ + gl) * H_ + hc + d0) = r;
}

// ---------------------------------------------------------------------------
extern "C" void kernel_launch(void* const* d_in, const int* in_sizes, int n_in,
                              void* d_out, int out_size, void* d_ws,
                              size_t ws_size, hipStream_t stream) {
  const float* queries = (const float*)d_in[0];
  const float* keys    = (const float*)d_in[1];
  const float* tK      = (const float*)d_in[2];
  const float* tV      = (const float*)d_in[3];
  const float* pK      = (const float*)d_in[4];
  const float* pV      = (const float*)d_in[5];
  const float* Wq      = (const float*)d_in[6];
  const float* bq      = (const float*)d_in[7];
  const float* Wk      = (const float*)d_in[8];
  const float* bk      = (const float*)d_in[9];
  const float* Wv      = (const float*)d_in[10];
  const float* bv      = (const float*)d_in[11];
  const unsigned char* tmask = (const unsigned char*)d_in[12];
  // d_in[13] (attn_mask) is the static causal triu; computed in-kernel.

  float* outp = (float*)d_out;
  float* Qb = (float*)d_ws;                       // B*L*H floats
  float* Kb = Qb + (size_t)B_ * L_ * H_;          // Keff = K + abs_pos_K
  float* Vb = Kb + (size_t)B_ * L_ * H_;          // Veff = V + abs_pos_V

  // 3 GEMMs * (2048/16)*(128/16) = 3072 tiles, 4 waves/block -> 768 blocks.
  proj_kernel<<<768, 128, 0, stream>>>(queries, keys, Wq, bq, Wk, bk, Wv, bv,
                                       pK, pV, Qb, Kb, Vb);
  // B * NH * (L/16) = 512 blocks.
  attn_kernel<<<512, 128, 0, stream>>>(Qb, Kb, Vb, tK, tV, tmask, outp);
}